// NoisyTopKGating_86165633893003
// MI455X (gfx1250) — compile-verified
//
#include <hip/hip_runtime.h>

typedef __attribute__((ext_vector_type(2))) float v2f;
typedef __attribute__((ext_vector_type(8))) float v8f;

static constexpr int kH        = 1024;  // hidden dim
static constexpr int kE        = 64;    // experts
static constexpr int kTopK     = 8;
static constexpr int kTokPerWG = 128;   // 4 waves x 32 tokens
static constexpr int kStride   = 65;    // LDS row stride (floats) to avoid bank conflicts

// Fused: router GEMM (fp32 WMMA) + softmax + top-8 + per-expert stats.
__global__ __launch_bounds__(kTokPerWG) void moe_router_kernel(
    const float* __restrict__ hs,      // [Ntok, 1024]
    const float* __restrict__ wt,      // [64, 1024]
    float* __restrict__ idx_out,       // [Ntok, 8] (indices as float)
    float* __restrict__ w_out,         // [Ntok, 8]
    float* __restrict__ wsS,           // [64] sum of softmax probs
    float* __restrict__ wsC)           // [64] top-k counts
{
    __shared__ float smem[kTokPerWG * kStride];  // logits -> probabilities
    __shared__ float cnt[kE];

    const int t     = threadIdx.x;
    const int lane  = t & 31;
    const int wave  = t >> 5;
    const int rowm  = lane & 15;          // M (A) / N (B) within tile
    const int halfk = (lane >> 4) * 2;    // K sub-offset: lanes 0-15 -> K 0,1 ; 16-31 -> K 2,3

    if (t < kE) cnt[t] = 0.0f;

    // ---------------- Phase 1: GEMM, 32 tokens x 64 experts per wave ----------------
    const long tok0 = (long)blockIdx.x * kTokPerWG + wave * 32;
    const float* A0 = hs + (tok0 + rowm) * kH;   // M-tile 0 row
    const float* A1 = A0 + 16 * kH;              // M-tile 1 row
    const float* Bb = wt + (long)rowm * kH + halfk;  // expert row base (N = rowm within tile)

    v8f acc[2][4];
#pragma unroll
    for (int mt = 0; mt < 2; ++mt)
#pragma unroll
        for (int nt = 0; nt < 4; ++nt)
            acc[mt][nt] = (v8f)(0.0f);

    for (int k0 = 0; k0 < kH; k0 += 4) {
        v2f a0 = *(const v2f*)(A0 + k0 + halfk);
        v2f a1 = *(const v2f*)(A1 + k0 + halfk);
        v2f b[4];
#pragma unroll
        for (int nt = 0; nt < 4; ++nt)
            b[nt] = *(const v2f*)(Bb + (long)nt * 16 * kH + k0);
#pragma unroll
        for (int nt = 0; nt < 4; ++nt) {
            acc[0][nt] = __builtin_amdgcn_wmma_f32_16x16x4_f32(
                false, a0, false, b[nt], (short)0, acc[0][nt], false, false);
            acc[1][nt] = __builtin_amdgcn_wmma_f32_16x16x4_f32(
                false, a1, false, b[nt], (short)0, acc[1][nt], false, false);
        }
    }

    // Scatter logits to LDS. C/D layout: VGPR r, lanes 0-15 -> M=r, lanes 16-31 -> M=r+8.
    const int trow = wave * 32 + (lane >> 4) * 8;
#pragma unroll
    for (int mt = 0; mt < 2; ++mt)
#pragma unroll
        for (int nt = 0; nt < 4; ++nt)
#pragma unroll
            for (int r = 0; r < 8; ++r)
                smem[(trow + mt * 16 + r) * kStride + nt * 16 + rowm] = acc[mt][nt][r];

    __syncthreads();

    // ---------------- Phase 2a: softmax per token (one thread per token) ----------------
    float* row = &smem[t * kStride];
    float mx = -3.4e38f;
    for (int e = 0; e < kE; ++e) mx = fmaxf(mx, row[e]);
    float s = 0.0f;
    for (int e = 0; e < kE; ++e) { float pe = __expf(row[e] - mx); s += pe; row[e] = pe; }
    float inv = 1.0f / s;
    for (int e = 0; e < kE; ++e) row[e] *= inv;

    __syncthreads();

    // ---------------- Phase 2b: per-expert prob column sums -> global ----------------
    if (t < kE) {
        float cs = 0.0f;
        for (int r = 0; r < kTokPerWG; ++r) cs += smem[r * kStride + t];
        atomicAdd(&wsS[t], cs);
    }
    __syncthreads();  // protect rows until all column reads done

    // ---------------- Phase 2c: destructive top-8 on own LDS row ----------------
    float topv[kTopK];
    int   topi[kTopK];
    float tsum = 0.0f;
#pragma unroll
    for (int j = 0; j < kTopK; ++j) {
        float best = -1.0f; int bi = 0;
        for (int e = 0; e < kE; ++e) {
            float v = row[e];
            if (v > best) { best = v; bi = e; }   // strict > : first index wins ties (stable)
        }
        row[bi] = -1.0f;
        topv[j] = best; topi[j] = bi; tsum += best;
        atomicAdd(&cnt[bi], 1.0f);
    }
    const float invs = 1.0f / (tsum + 1e-20f);
    const long gt = (long)blockIdx.x * kTokPerWG + t;
#pragma unroll
    for (int j = 0; j < kTopK; ++j) {
        idx_out[gt * kTopK + j] = (float)topi[j];
        w_out[gt * kTopK + j]   = topv[j] * invs;
    }
    __syncthreads();

    // ---------------- Phase 2d: counts -> global ----------------
    if (t < kE) atomicAdd(&wsC[t], cnt[t]);
}

// aux_loss = ALPHA * E / (N*N*K) * sum_e S_e * C_e
__global__ void moe_aux_kernel(const float* __restrict__ wsS,
                               const float* __restrict__ wsC,
                               float* __restrict__ aux, int Ntok)
{
    if (threadIdx.x == 0 && blockIdx.x == 0) {
        float s = 0.0f;
        for (int e = 0; e < kE; ++e) s += wsS[e] * wsC[e];
        double scale = 0.01 * (double)kE /
                       ((double)Ntok * (double)Ntok * (double)kTopK);
        aux[0] = (float)((double)s * scale);
    }
}

extern "C" void kernel_launch(void* const* d_in, const int* in_sizes, int n_in,
                              void* d_out, int out_size, void* d_ws, size_t ws_size,
                              hipStream_t stream)
{
    const float* hs = (const float*)d_in[0];   // [4,8192,1024] fp32
    const float* wt = (const float*)d_in[1];   // [64,1024] fp32
    float* out = (float*)d_out;

    const int Ntok = in_sizes[0] / kH;         // 32768
    float* idx_out = out;
    float* w_out   = out + (long)Ntok * kTopK;
    float* aux     = out + 2L * Ntok * kTopK;

    float* wsS = (float*)d_ws;
    float* wsC = wsS + kE;

    hipMemsetAsync(d_ws, 0, 2 * kE * sizeof(float), stream);

    const int nBlocks = Ntok / kTokPerWG;
    moe_router_kernel<<<nBlocks, kTokPerWG, 0, stream>>>(hs, wt, idx_out, w_out, wsS, wsC);
    moe_aux_kernel<<<1, 1, 0, stream>>>(wsS, wsC, aux, Ntok);
}